// WhisperPreprocessTVM_2456721293848
// MI455X (gfx1250) — compile-verified
//
#include <hip/hip_runtime.h>
#include <hip/hip_bf16.h>

typedef __attribute__((ext_vector_type(2))) float v2f;
typedef __attribute__((ext_vector_type(8))) float v8f;

#define N_SAMPLES   480000
#define N_FRAMES    3000
#define N_FFT       400
#define N_FREQ      201
#define N_MELS      80
#define HOP         160
#define BATCH       64
#define BASIS_COLS  416          // 0..200 real, 208..408 imag, rest zero
#define KPAD        208          // padded mel K (freq) dimension
#define LDSA_STRIDE 401          // odd -> no LDS bank conflicts
#define LDSP_STRIDE 209          // odd -> no LDS bank conflicts
#define MTILE       32           // frames per workgroup (2 WMMA m-tiles)
#define PI_D        3.14159265358979323846

// ---- order-preserving float<->uint encoding for atomic max ----
__device__ __forceinline__ unsigned enc_f32(float x) {
    unsigned u = __float_as_uint(x);
    return (u & 0x80000000u) ? ~u : (u | 0x80000000u);
}
__device__ __forceinline__ float dec_f32(unsigned e) {
    return (e & 0x80000000u) ? __uint_as_float(e ^ 0x80000000u) : __uint_as_float(~e);
}

// ---------------------------------------------------------------
// Kernel 1: build windowed DFT basis [400][416], zero-padded mel
//           weights [208][80], and init per-batch max buffer
// ---------------------------------------------------------------
__global__ void k_setup(float* __restrict__ basis, float* __restrict__ melpad,
                        const float* __restrict__ melw, unsigned* __restrict__ maxbuf) {
    int i = blockIdx.x * blockDim.x + threadIdx.x;
    if (i < BATCH) maxbuf[i] = 0u;   // encodes below any real value
    if (i < KPAD * N_MELS) {
        int k = i / N_MELS;
        int m = i - k * N_MELS;
        melpad[i] = (k < N_FREQ) ? melw[k * N_MELS + m] : 0.0f;
    }
    if (i >= N_FFT * BASIS_COLS) return;
    int k = i / BASIS_COLS;
    int n = i - k * BASIS_COLS;
    double w = 0.5 - 0.5 * cos(2.0 * PI_D * (double)k / (double)N_FFT);
    float v = 0.0f;
    if (n < N_FREQ) {
        v = (float)(cos(2.0 * PI_D / (double)N_FFT * (double)n * (double)k) * w);
    } else if (n >= 208 && n < 208 + N_FREQ) {
        int f = n - 208;
        v = (float)(-sin(2.0 * PI_D / (double)N_FFT * (double)f * (double)k) * w);
    }
    basis[i] = v;
}

// ---------------------------------------------------------------
// Kernel 2: fused STFT(WMMA f32) + power + mel(WMMA f32) + log10
//           one workgroup = 32 frames of one batch, 13 waves
// ---------------------------------------------------------------
__device__ __forceinline__ v8f wmma_f32(v2f a, v2f b, v8f c) {
    return __builtin_amdgcn_wmma_f32_16x16x4_f32(
        /*neg_a=*/false, a, /*neg_b=*/false, b,
        /*c_mod=*/(short)0, c, /*reuse_a=*/false, /*reuse_b=*/false);
}

__global__ __launch_bounds__(416) void k_fused(const float* __restrict__ wave,
                                               const float* __restrict__ melpad,
                                               const float* __restrict__ basis,
                                               float* __restrict__ out,
                                               unsigned* __restrict__ maxbuf) {
    __shared__ float ldsA[MTILE * LDSA_STRIDE];   // 32 frames x 400 taps (stride 401)
    __shared__ float ldsP[MTILE * LDSP_STRIDE];   // power: 32 frames x 208 freqs
    __shared__ float ldsM[N_MELS * MTILE];        // log-mel tile [mel][t]

    const int b  = blockIdx.y;
    const int t0 = blockIdx.x * MTILE;
    const float* wv = wave + (size_t)b * N_SAMPLES;

    // ---- stage 32 overlapping frames into LDS with reflect padding ----
    for (int i = threadIdx.x; i < MTILE * N_FFT; i += 416) {
        int fr = i / N_FFT;
        int j  = i - fr * N_FFT;
        int q  = (t0 + fr) * HOP + j - 200;                      // reflect-pad index
        int idx = q < 0 ? -q : (q > N_SAMPLES - 1 ? 2 * (N_SAMPLES - 1) - q : q);
        ldsA[fr * LDSA_STRIDE + j] = wv[idx];
    }
    __syncthreads();

    const int wave_id = threadIdx.x >> 5;
    const int lane    = threadIdx.x & 31;
    const int mrow    = lane & 15;             // A: M row / B: N col within tile
    const int ksel    = (lane >> 4) << 1;      // lanes 0-15: K+0/1, lanes 16-31: K+2/3
    const int rhi     = (lane >= 16) ? 8 : 0;  // D: upper-lane rows +8

    // ---- STFT: wave w owns freq pair-tile w (real cols 16w, imag cols 208+16w),
    //      computed for both 16-frame m-tiles so B fragments are reused 2x ----
    {
        const int c_re = wave_id * 16;
        const int c_im = 208 + wave_id * 16;
        v8f ar0 = {}, ai0 = {}, ar1 = {}, ai1 = {};
        const int rowA0 = mrow * LDSA_STRIDE;
        const int rowA1 = (16 + mrow) * LDSA_STRIDE;
        #pragma unroll 2
        for (int k0 = 0; k0 < N_FFT; k0 += 4) {
            const int k = k0 + ksel;
            v2f br; br.x = basis[k * BASIS_COLS + c_re + mrow];
                    br.y = basis[(k + 1) * BASIS_COLS + c_re + mrow];
            v2f bi; bi.x = basis[k * BASIS_COLS + c_im + mrow];
                    bi.y = basis[(k + 1) * BASIS_COLS + c_im + mrow];
            v2f a0; a0.x = ldsA[rowA0 + k]; a0.y = ldsA[rowA0 + k + 1];
            v2f a1; a1.x = ldsA[rowA1 + k]; a1.y = ldsA[rowA1 + k + 1];
            ar0 = wmma_f32(a0, br, ar0);
            ai0 = wmma_f32(a0, bi, ai0);
            ar1 = wmma_f32(a1, br, ar1);
            ai1 = wmma_f32(a1, bi, ai1);
        }
        const int c = wave_id * 16 + mrow;     // freq col (201..207 stay zero)
        #pragma unroll
        for (int j = 0; j < 8; ++j) {
            int m = j + rhi;                   // frame row within m-tile
            float r0 = ar0[j], i0 = ai0[j];
            float r1 = ar1[j], i1 = ai1[j];
            ldsP[m * LDSP_STRIDE + c]        = r0 * r0 + i0 * i0;
            ldsP[(16 + m) * LDSP_STRIDE + c] = r1 * r1 + i1 * i1;
        }
    }
    __syncthreads();

    // ---- mel GEMM: waves 0..9 -> (m-tile, 16-mel n-tile); unconditional loads ----
    if (wave_id < 10) {
        const int mt = (wave_id >= 5) ? 1 : 0;
        const int n0 = (wave_id - mt * 5) * 16;
        const int rowP = (mt * 16 + mrow) * LDSP_STRIDE;
        v8f acc = {};
        #pragma unroll 4
        for (int k0 = 0; k0 < KPAD; k0 += 4) {
            const int k = k0 + ksel;
            v2f a;  a.x  = ldsP[rowP + k];
                    a.y  = ldsP[rowP + k + 1];
            v2f bm; bm.x = melpad[k * N_MELS + n0 + mrow];
                    bm.y = melpad[(k + 1) * N_MELS + n0 + mrow];
            acc = wmma_f32(a, bm, acc);
        }
        float vmax = -3.0e38f;
        #pragma unroll
        for (int j = 0; j < 8; ++j) {
            int trow = mt * 16 + j + rhi;
            float x = acc[j];
            float y = __log2f(fmaxf(x, 1e-10f)) * 0.30102999566398120f;  // log10
            ldsM[(n0 + mrow) * MTILE + trow] = y;
            if (t0 + trow < N_FRAMES) vmax = fmaxf(vmax, y);
        }
        #pragma unroll
        for (int off = 16; off > 0; off >>= 1)
            vmax = fmaxf(vmax, __shfl_xor(vmax, off, 32));
        if (lane == 0) atomicMax(maxbuf + b, enc_f32(vmax));
    }
    __syncthreads();

    // ---- coalesced store of unnormalized log-mel into d_out [b][mel][t] ----
    for (int i = threadIdx.x; i < N_MELS * MTILE; i += 416) {
        int m  = i >> 5;
        int tl = i & 31;
        int t  = t0 + tl;
        if (t < N_FRAMES)
            out[(size_t)b * (N_MELS * N_FRAMES) + m * N_FRAMES + t] = ldsM[i];
    }
}

// ---------------------------------------------------------------
// Kernel 3: clamp to (max-8) and scale, in place on d_out
// ---------------------------------------------------------------
__global__ void k_final(float* __restrict__ out, const unsigned* __restrict__ maxbuf) {
    int i = blockIdx.x * blockDim.x + threadIdx.x;
    if (i >= BATCH * N_MELS * N_FRAMES) return;
    int b = i / (N_MELS * N_FRAMES);
    float thr = dec_f32(maxbuf[b]) - 8.0f;
    float x = out[i];
    out[i] = (fmaxf(x, thr) + 4.0f) * 0.25f;
}

// ---------------------------------------------------------------
// Kernel 4: attention mask (int32 written into tail of d_out)
// ---------------------------------------------------------------
__global__ void k_mask(int* __restrict__ outmask, const int* __restrict__ valid) {
    int i = blockIdx.x * blockDim.x + threadIdx.x;
    if (i >= BATCH * N_FRAMES) return;
    int b = i / N_FRAMES;
    int t = i - b * N_FRAMES;
    outmask[i] = (t * HOP < valid[b]) ? 1 : 0;
}

extern "C" void kernel_launch(void* const* d_in, const int* in_sizes, int n_in,
                              void* d_out, int out_size, void* d_ws, size_t ws_size,
                              hipStream_t stream) {
    const float* waveform = (const float*)d_in[0];
    const int*   valid    = (const int*)d_in[1];
    const float* melw     = (const float*)d_in[2];

    float* out_feat = (float*)d_out;                                   // [64,80,3000]
    int*   out_mask = (int*)d_out + BATCH * N_MELS * N_FRAMES;         // [64,3000]

    // workspace layout
    float*    basis  = (float*)d_ws;                                   // 400*416 f32
    float*    melpad = (float*)((char*)d_ws + (size_t)N_FFT * BASIS_COLS * sizeof(float));
    unsigned* maxbuf = (unsigned*)((char*)melpad + (size_t)KPAD * N_MELS * sizeof(float));

    // 1) basis + padded mel weights + max init
    {
        int n = N_FFT * BASIS_COLS;
        k_setup<<<(n + 255) / 256, 256, 0, stream>>>(basis, melpad, melw, maxbuf);
    }
    // 2) fused STFT + mel + log10 (WMMA f32)
    {
        dim3 grid((N_FRAMES + MTILE - 1) / MTILE, BATCH);   // 94 x 64 tiles
        k_fused<<<grid, 416, 0, stream>>>(waveform, melpad, basis, out_feat, maxbuf);
    }
    // 3) normalize in place
    {
        int n = BATCH * N_MELS * N_FRAMES;
        k_final<<<(n + 255) / 256, 256, 0, stream>>>(out_feat, maxbuf);
    }
    // 4) attention mask
    {
        int n = BATCH * N_FRAMES;
        k_mask<<<(n + 255) / 256, 256, 0, stream>>>(out_mask, valid);
    }
}